// MoEFFN_81862076662211
// MI455X (gfx1250) — compile-verified
//
#include <hip/hip_runtime.h>
#include <math.h>
#include <stdint.h>

// ---------------- problem constants ----------------
#define D_MODEL   1024
#define D_EXPERT  2048
#define N_EXP     8
#define NTOK      4096      // 2 * 2048 tokens
#define TILE_M    32        // token rows per block (two WMMA M-tiles)
#define CHUNK     256       // expert-dim columns per phase pair (8 chunks)
#define NCHUNK    (D_EXPERT / CHUNK)
#define THREADS   512       // 16 waves
#define SMEM_BYTES ((TILE_M * D_MODEL + TILE_M * CHUNK) * 4)   // 160KB exactly

typedef float v2f __attribute__((ext_vector_type(2)));
typedef float v8f __attribute__((ext_vector_type(8)));

// D = A(16x4 f32) * B(4x16 f32) + C(16x16 f32), wave32 striped.
__device__ __forceinline__ v8f wmma_f32(v2f a, v2f b, v8f c) {
  return __builtin_amdgcn_wmma_f32_16x16x4_f32(false, a, false, b, (short)0, c,
                                               false, false);
}

// CDNA5 async copy: copies 16 bytes from the global address gaddr to the LDS address lds_off, tracked by ASYNCcnt.
__device__ __forceinline__ void async_to_lds_b128(uint32_t lds_off,
                                                  const float* gaddr) {
  asm volatile("global_load_async_to_lds_b128 %0, %1, off"
               :: "v"(lds_off), "v"(gaddr)
               : "memory");
}
__device__ __forceinline__ void wait_async0() {
  asm volatile("s_wait_asynccnt 0" ::: "memory");
}

// XOR bank swizzle (touches addr bits [5:2] only): conflict-free column reads
// while keeping 4-float groups contiguous (b64/b128 friendly, no padding).
__device__ __forceinline__ int xsw(int row, int col) {
  return row * D_MODEL + (col ^ ((row & 15) * 4));
}
__device__ __forceinline__ int asw(int row, int col) {
  return row * CHUNK + (col ^ ((row & 15) * 4));
}

__device__ __forceinline__ v2f ld2(const float* p) {          // contiguous pair
  v2f r; r.x = p[0]; r.y = p[1]; return r;
}
__device__ __forceinline__ v2f ldB(const float* p, int str) { // row-strided pair
  v2f r; r.x = p[0]; r.y = p[str]; return r;
}

// ---------------- kernel 1: router + top-2 gather ----------------
__global__ void router_topk(const float* __restrict__ x,
                            const float* __restrict__ rW,
                            int* __restrict__ counts,
                            int* __restrict__ lists,
                            float* __restrict__ gws) {
  const int lane = threadIdx.x & 31;
  const int wave = threadIdx.x >> 5;
  const int tok  = blockIdx.x * 8 + wave;
  if (tok >= NTOK) return;

  float acc[N_EXP];
#pragma unroll
  for (int e = 0; e < N_EXP; ++e) acc[e] = 0.0f;

  const float* xr = x + (size_t)tok * D_MODEL;
  for (int i = lane; i < D_MODEL; i += 32) {
    float xv = xr[i];
    const float4* w4 = (const float4*)(rW + (size_t)i * N_EXP);
    float4 w0 = w4[0], w1 = w4[1];
    acc[0] += xv * w0.x; acc[1] += xv * w0.y;
    acc[2] += xv * w0.z; acc[3] += xv * w0.w;
    acc[4] += xv * w1.x; acc[5] += xv * w1.y;
    acc[6] += xv * w1.z; acc[7] += xv * w1.w;
  }
#pragma unroll
  for (int e = 0; e < N_EXP; ++e) {
#pragma unroll
    for (int m = 16; m >= 1; m >>= 1)
      acc[e] += __shfl_xor(acc[e], m, 32);
  }

  if (lane == 0) {
    float mx = acc[0];
#pragma unroll
    for (int e = 1; e < N_EXP; ++e) mx = fmaxf(mx, acc[e]);
    float p[N_EXP], s = 0.0f;
#pragma unroll
    for (int e = 0; e < N_EXP; ++e) { p[e] = __expf(acc[e] - mx); s += p[e]; }
    float inv = 1.0f / s;

    int i1 = 0;
#pragma unroll
    for (int e = 1; e < N_EXP; ++e) if (p[e] > p[i1]) i1 = e;
    int i2 = (i1 == 0) ? 1 : 0;
#pragma unroll
    for (int e = 0; e < N_EXP; ++e)
      if (e != i1 && p[e] > p[i2]) i2 = e;

    float v1 = p[i1] * inv, v2 = p[i2] * inv;
    float dn = 1.0f / (v1 + v2 + 1e-8f);

    int pos1 = atomicAdd(&counts[i1], 1);
    lists[i1 * NTOK + pos1] = tok;
    gws[i1 * NTOK + pos1]   = v1 * dn;
    int pos2 = atomicAdd(&counts[i2], 1);
    lists[i2 * NTOK + pos2] = tok;
    gws[i2 * NTOK + pos2]   = v2 * dn;
  }
}

// row -> (token, combine weight); pad rows get weight 0 (contribute +0.0)
__device__ __forceinline__ void row_info(int routed, int cnt,
                                         const int* __restrict__ lst,
                                         const float* __restrict__ gw,
                                         int idx, int* tok, float* w) {
  if (routed) {
    int ic = idx < cnt ? idx : cnt - 1;
    *tok = lst[ic];
    *w   = idx < cnt ? gw[ic] : 0.0f;
  } else {
    *tok = idx; *w = 1.0f;
  }
}

// ---------------- kernel 2: fused SwiGLU expert (shared + routed) ----------------
// Block = 32-token tile x expert, 512 threads = 16 waves.
// M=32: every B (weight) fragment feeds two M-tiles -> half the L2 weight
// traffic of M=16. Inner loops are software-pipelined: iteration i issues the
// loads for i+1 BEFORE executing i's four WMMAs, hiding L2 latency behind the
// matrix pipe. Combine weight applied at the epilogue (reference ordering).
__global__ void moe_ffn(const float* __restrict__ x,
                        const float* __restrict__ Wup_all,
                        const float* __restrict__ Wgate_all,
                        const float* __restrict__ Wdown_all,
                        const int* __restrict__ counts,
                        const int* __restrict__ lists,
                        const float* __restrict__ gws,
                        float* __restrict__ out,
                        int routed) {
  extern __shared__ float smem[];
  float* Xs = smem;                        // [TILE_M][D_MODEL], swizzled
  float* Ac = smem + TILE_M * D_MODEL;     // [TILE_M][CHUNK],  swizzled

  const int tile = blockIdx.x;
  const int e    = blockIdx.y;

  const float* up   = Wup_all;
  const float* gate = Wgate_all;
  const float* down = Wdown_all;
  const int* lst  = lists + e * NTOK;
  const float* gw = gws   + e * NTOK;
  int cnt = NTOK;

  if (routed) {
    cnt = counts[e];
    if (tile * TILE_M >= cnt) return;                // early-exit empty tiles
    up   += (size_t)e * D_MODEL * D_EXPERT;
    gate += (size_t)e * D_MODEL * D_EXPERT;
    down += (size_t)e * D_EXPERT * D_MODEL;
  }

  // Stage X tile via async global->LDS DMA: 16 threads/row, 64 floats each.
  {
    int r  = threadIdx.x >> 4;                       // 0..31
    int c0 = (threadIdx.x & 15) * (D_MODEL / 16);    // 64 floats
    int tok; float wunused;
    row_info(routed, cnt, lst, gw, tile * TILE_M + r, &tok, &wunused);
    const float* src = x + (size_t)tok * D_MODEL + c0;
#pragma unroll
    for (int i = 0; i < D_MODEL / 16; i += 4)
      async_to_lds_b128((uint32_t)(uintptr_t)(Xs + xsw(r, c0 + i)), src + i);
  }
  wait_async0();
  __syncthreads();

  const int lane = threadIdx.x & 31;
  const int wv   = threadIdx.x >> 5;       // 0..15
  const int hi   = lane >> 4;              // half-wave: K pair (A/B), +8 rows (C/D)
  const int ln   = lane & 15;

  v8f oacc[8];                             // [4*tp + {nt0_lo,nt0_hi,nt1_lo,nt1_hi}]
#pragma unroll
  for (int t = 0; t < 8; ++t) oacc[t] = (v8f){};

  for (int c = 0; c < NCHUNK; ++c) {
    // ---- phase 1: wave wv owns chunk cols [wv*16, wv*16+16) ----
    {
      const int lc = wv * 16 + ln;
      const int gc = c * CHUNK + lc;
      v8f au0 = (v8f){}, ag0 = (v8f){};    // rows 0..15
      v8f au1 = (v8f){}, ag1 = (v8f){};    // rows 16..31

      // pipeline prologue: fragments for k=0
      v2f a0 = ld2(Xs + xsw(ln,      2 * hi));
      v2f a1 = ld2(Xs + xsw(ln + 16, 2 * hi));
      v2f bu = ldB(up   + (size_t)(2 * hi) * D_EXPERT + gc, D_EXPERT);
      v2f bg = ldB(gate + (size_t)(2 * hi) * D_EXPERT + gc, D_EXPERT);

      for (int k = 0; k < D_MODEL; k += 4) {
        const int kn = (k + 4) & (D_MODEL - 1);      // wraps to 0 on last iter
        // issue next iteration's loads first (overlap with WMMAs below)
        v2f a0n = ld2(Xs + xsw(ln,      kn + 2 * hi));
        v2f a1n = ld2(Xs + xsw(ln + 16, kn + 2 * hi));
        v2f bun = ldB(up   + (size_t)(kn + 2 * hi) * D_EXPERT + gc, D_EXPERT);
        v2f bgn = ldB(gate + (size_t)(kn + 2 * hi) * D_EXPERT + gc, D_EXPERT);

        au0 = wmma_f32(a0, bu, au0);       // B reused across both M-tiles
        au1 = wmma_f32(a1, bu, au1);
        ag0 = wmma_f32(a0, bg, ag0);
        ag1 = wmma_f32(a1, bg, ag1);

        a0 = a0n; a1 = a1n; bu = bun; bg = bgn;
      }
#pragma unroll
      for (int r = 0; r < 8; ++r) {
        const int row0 = r + 8 * hi;       // C/D row mapping
        const int row1 = row0 + 16;
        float u0 = au0[r], g0 = ag0[r];
        float u1 = au1[r], g1 = ag1[r];
        Ac[asw(row0, lc)] = (u0 / (1.0f + __expf(-u0))) * g0;
        Ac[asw(row1, lc)] = (u1 / (1.0f + __expf(-u1))) * g1;
      }
    }
    __syncthreads();

    // ---- phase 2: wave wv owns out cols [wv*64,+64): 2 pairs of N-tiles ----
#pragma unroll
    for (int tp = 0; tp < 2; ++tp) {
      const int oc0 = wv * 64 + tp * 32 + ln;
      v8f d00 = oacc[4 * tp + 0], d01 = oacc[4 * tp + 1];  // N+0, rows lo/hi
      v8f d10 = oacc[4 * tp + 2], d11 = oacc[4 * tp + 3];  // N+16, rows lo/hi

      // pipeline prologue: fragments for kk=0
      v2f a0 = ld2(Ac + asw(ln,      2 * hi));
      v2f a1 = ld2(Ac + asw(ln + 16, 2 * hi));
      const float* pd0 = down + (size_t)(c * CHUNK + 2 * hi) * D_MODEL + oc0;
      v2f b0 = ldB(pd0,      D_MODEL);
      v2f b1 = ldB(pd0 + 16, D_MODEL);

      for (int kk = 0; kk < CHUNK; kk += 4) {
        const int kn = (kk + 4) & (CHUNK - 1);       // wraps to 0 on last iter
        v2f a0n = ld2(Ac + asw(ln,      kn + 2 * hi));
        v2f a1n = ld2(Ac + asw(ln + 16, kn + 2 * hi));
        const float* pdn =
            down + (size_t)(c * CHUNK + kn + 2 * hi) * D_MODEL + oc0;
        v2f b0n = ldB(pdn,      D_MODEL);
        v2f b1n = ldB(pdn + 16, D_MODEL);

        d00 = wmma_f32(a0, b0, d00);       // B reused across M-tiles
        d01 = wmma_f32(a1, b0, d01);
        d10 = wmma_f32(a0, b1, d10);
        d11 = wmma_f32(a1, b1, d11);

        a0 = a0n; a1 = a1n; b0 = b0n; b1 = b1n;
      }
      oacc[4 * tp + 0] = d00; oacc[4 * tp + 1] = d01;
      oacc[4 * tp + 2] = d10; oacc[4 * tp + 3] = d11;
    }
    __syncthreads();
  }

  // ---- epilogue: scale by combine weight, store (shared) / scatter-add (routed)
#pragma unroll
  for (int tp = 0; tp < 2; ++tp) {
    const int oc0 = wv * 64 + tp * 32 + ln;
#pragma unroll
    for (int r = 0; r < 8; ++r) {
      const int row0 = r + 8 * hi;
      const int row1 = row0 + 16;
      int tok0, tok1; float w0, w1;
      row_info(routed, cnt, lst, gw, tile * TILE_M + row0, &tok0, &w0);
      row_info(routed, cnt, lst, gw, tile * TILE_M + row1, &tok1, &w1);
      float* dst0 = out + (size_t)tok0 * D_MODEL + oc0;
      float* dst1 = out + (size_t)tok1 * D_MODEL + oc0;
      float v00 = oacc[4 * tp + 0][r] * w0, v01 = oacc[4 * tp + 1][r] * w1;
      float v10 = oacc[4 * tp + 2][r] * w0, v11 = oacc[4 * tp + 3][r] * w1;
      if (routed) {
        atomicAdd(dst0, v00);      atomicAdd(dst1, v01);
        atomicAdd(dst0 + 16, v10); atomicAdd(dst1 + 16, v11);
      } else {
        dst0[0] = v00;  dst1[0] = v01;
        dst0[16] = v10; dst1[16] = v11;
      }
    }
  }
}

// ---------------- host launcher ----------------
extern "C" void kernel_launch(void* const* d_in, const int* in_sizes, int n_in,
                              void* d_out, int out_size, void* d_ws, size_t ws_size,
                              hipStream_t stream) {
  const float* x      = (const float*)d_in[0];
  const float* sWup   = (const float*)d_in[1];
  const float* sWgate = (const float*)d_in[2];
  const float* sWdown = (const float*)d_in[3];
  const float* rWup   = (const float*)d_in[4];
  const float* rWgate = (const float*)d_in[5];
  const float* rWdown = (const float*)d_in[6];
  const float* routerW= (const float*)d_in[7];
  float* out = (float*)d_out;

  // workspace: [counts: 8 ints, padded to 256B][lists: 8*4096 int][gws: 8*4096 f32]
  int*   counts = (int*)d_ws;
  int*   lists  = (int*)((char*)d_ws + 256);
  float* gws    = (float*)((char*)d_ws + 256 + (size_t)N_EXP * NTOK * sizeof(int));

  hipMemsetAsync(d_ws, 0, 256, stream);   // zero per-expert counters

  (void)hipFuncSetAttribute((const void*)moe_ffn,
                            hipFuncAttributeMaxDynamicSharedMemorySize,
                            SMEM_BYTES);

  // 1) router + top-2 gather
  router_topk<<<NTOK / 8, 256, 0, stream>>>(x, routerW, counts, lists, gws);

  // 2) shared expert: plain stores initialize d_out fully
  moe_ffn<<<dim3(NTOK / TILE_M, 1), THREADS, SMEM_BYTES, stream>>>(
      x, sWup, sWgate, sWdown, counts, lists, gws, out, 0);

  // 3) routed experts (expert = slow axis -> 25MB/expert L2-resident),
  //    scatter-add on top of the shared result
  moe_ffn<<<dim3(NTOK / TILE_M, N_EXP), THREADS, SMEM_BYTES, stream>>>(
      x, rWup, rWgate, rWdown, counts, lists, gws, out, 1);
}